// CapsRoutingLayer_77549929496796
// MI455X (gfx1250) — compile-verified
//
#include <hip/hip_runtime.h>
#include <math.h>

// CapsNet dynamic routing on MI455X (gfx1250, wave32).
// x_hat (604MB) is never materialized; it is recomputed per routing iteration
// with V_WMMA_F32_16X16X4_F32. W (151MB) + x (19MB) fit in the 192MB L2, so
// recompute streams from L2 and the pass is matrix-f32 throughput bound.
//
// K-partition trick: WMMA K-step kk covers global k = {2kk,2kk+1} (lanes 0-15,
// HW K0/K1) and {16+2kk,16+2kk+1} (lanes 16-31, HW K2/K3). A lane's 8 B (and A)
// operands are then 16 CONTIGUOUS floats -> 4 x b128 loads into distinct
// registers, so loads clause and overlap the WMMA chain instead of serializing.
//
// Round-3 fix: the unconstrained scheduler over-hoisted B loads (18-deep
// clause), ran out of VGPRs, and the tail tiles collapsed back to
// load->wait(0)->wmma serialization. Pin the schedule with sched_group_barrier
// to a steady two-tile-deep pipeline: [4 DS A][8 VMEM B] ([8 WMMA][4 VMEM B])x6
// [8 WMMA][8 WMMA] -> at most 3 B tiles (48 VGPRs) live, no tail collapse.

typedef float v2f __attribute__((ext_vector_type(2)));
typedef float v4f __attribute__((ext_vector_type(4)));
typedef float v8f __attribute__((ext_vector_type(8)));

#define CAPS_B    128
#define CAPS_NI   1152
#define CAPS_DI   32
#define CAPS_NO   32
#define CAPS_OD   1024                    // NO * DO
#define I_CHUNK   16
#define N_CHUNKS  (CAPS_NI / I_CHUNK)     // 72
#define BT        16                      // batch tile (WMMA M)
#define NBT       (CAPS_B / BT)           // 8
#define S_ELEMS   (CAPS_B * CAPS_OD)      // 131072
#define XT_PITCH  36                      // floats; 144B rows -> b128-aligned

// sched_group_barrier masks (LLVM AMDGPU): 0x8 MFMA/WMMA, 0x20 VMEM read,
// 0x100 DS read.
#define SGB(mask, cnt) __builtin_amdgcn_sched_group_barrier((mask), (cnt), 0)

__global__ __launch_bounds__(256) void caps_zero_norms(float* norms) {
  if (threadIdx.x < 4) norms[threadIdx.x] = 0.0f;
}

// One fused pass. mode==0: initial uniform routing (c = 1/32, accumulate s).
// mode==1: logit update + softmax + coefficient-weighted s accumulation.
// Grid: (i-chunk, b-tile), 256 threads = 8 waves; wave w owns od columns
// [128w, 128w+128) = output capsules {4w..4w+3}.
__global__ __launch_bounds__(256)
void caps_pass(const float* __restrict__ x,     // [B][NI][DI]
               const float* __restrict__ W,     // [NI][OD][DI]
               const float* vin,                // [B][OD]  (no restrict: blocks LICM)
               float* __restrict__ blog,        // [B][NI][NO] routing logits
               float* __restrict__ s_part,      // [N_CHUNKS][B][OD]
               int mode, int use_bold)
{
  __shared__ float xt[BT][XT_PITCH];      // x slice, b128-aligned rows
  __shared__ float cb[BT][CAPS_NO + 1];   // logits -> exp() scratch
  __shared__ float rinv[BT];              // 1/rowsum for softmax

  const int tid  = threadIdx.x;
  const int wave = tid >> 5;
  const int lane = tid & 31;
  const int lh   = lane >> 4;             // half-wave select (HW K0K1 / K2K3)
  const int li   = lane & 15;

  const int ic   = blockIdx.x;            // i-chunk index
  const int b0   = blockIdx.y * BT;       // batch tile base
  const int col0 = wave * 128;            // od column base for this wave

  // s accumulator: 8 N-tiles x v8f, held in VGPRs across the whole i loop
  v8f sacc[8];
  #pragma unroll
  for (int nt = 0; nt < 8; ++nt)
    #pragma unroll
    for (int r = 0; r < 8; ++r) sacc[nt][r] = 0.0f;

  for (int ii = 0; ii < I_CHUNK; ++ii) {
    const int i = ic * I_CHUNK + ii;

    __syncthreads();   // protect xt reuse across iterations
    #pragma unroll
    for (int rep = 0; rep < 2; ++rep) {
      int e  = tid + rep * 256;           // 512 elements, 2 per thread
      int br = e >> 5, k = e & 31;
      xt[br][k] = x[((size_t)(b0 + br) * CAPS_NI + i) * CAPS_DI + k];
    }
    // warm next i's W slice in L2; lives in this scheduling region so it
    // doesn't perturb the pinned VMEM groups below the barrier.
    if (i + 1 < CAPS_NI)
      __builtin_prefetch(W + (size_t)(i + 1) * CAPS_OD * CAPS_DI +
                             (size_t)(col0 + lane * 4) * CAPS_DI, 0, 0);
    __syncthreads();

    // A operands: lane li = M row; this lane's 16 floats are contiguous
    // (k = 16*lh .. 16*lh+15) -> 4 x ds_load_b128 into distinct registers.
    v4f A4[4];
    #pragma unroll
    for (int q = 0; q < 4; ++q)
      A4[q] = *(const v4f*)&xt[li][16 * lh + 4 * q];
    const v2f* A = (const v2f*)A4;        // A[kk] = HW-K pair of step kk

    // GEMM: x_hat tile (16b x 128od), 8 N-tiles x 8 K-steps of WMMA.
    // Two interleaved accumulators per tile break the WMMA->WMMA RAW chain.
    const float* wbase = W + (size_t)i * CAPS_OD * CAPS_DI
                           + (size_t)(col0 + li) * CAPS_DI + 16 * lh;
    v8f xh[8];
    #pragma unroll
    for (int nt = 0; nt < 8; ++nt) {
      const float* wt = wbase + nt * (16 * CAPS_DI);   // +2048B per tile
      v4f B4[4];
      #pragma unroll
      for (int q = 0; q < 4; ++q)
        B4[q] = *(const v4f*)(wt + 4 * q);
      const v2f* Bv = (const v2f*)B4;
      v8f acc0, acc1;
      #pragma unroll
      for (int r = 0; r < 8; ++r) { acc0[r] = 0.0f; acc1[r] = 0.0f; }
      #pragma unroll
      for (int kk = 0; kk < 4; ++kk) {
        acc0 = __builtin_amdgcn_wmma_f32_16x16x4_f32(
            false, A[2 * kk],     false, Bv[2 * kk],
            (short)0, acc0, false, false);
        acc1 = __builtin_amdgcn_wmma_f32_16x16x4_f32(
            false, A[2 * kk + 1], false, Bv[2 * kk + 1],
            (short)0, acc1, false, false);
      }
      xh[nt] = acc0 + acc1;
    }

    // Pin the schedule: two-tile-deep software pipeline with bounded pressure.
    SGB(0x100, 4);                 // A: 4 ds_load_b128
    SGB(0x020, 8);                 // B loads: tiles 0,1
    SGB(0x008, 8); SGB(0x020, 4);  // wmma t0 | B t2
    SGB(0x008, 8); SGB(0x020, 4);  // wmma t1 | B t3
    SGB(0x008, 8); SGB(0x020, 4);  // wmma t2 | B t4
    SGB(0x008, 8); SGB(0x020, 4);  // wmma t3 | B t5
    SGB(0x008, 8); SGB(0x020, 4);  // wmma t4 | B t6
    SGB(0x008, 8); SGB(0x020, 4);  // wmma t5 | B t7
    SGB(0x008, 8);                 // wmma t6
    SGB(0x008, 8);                 // wmma t7

    if (mode == 0) {                        // iteration 0: c == 1/32 uniform
      #pragma unroll
      for (int nt = 0; nt < 8; ++nt)
        #pragma unroll
        for (int r = 0; r < 8; ++r)
          sacc[nt][r] += xh[nt][r] * (1.0f / 32.0f);
      continue;
    }

    // v in D-matrix register layout, reloaded per i (L0/L2-hot; keeps VGPR
    // count bounded instead of pinning 64 registers across the whole loop).
    v8f vreg[8];
    #pragma unroll
    for (int nt = 0; nt < 8; ++nt)
      #pragma unroll
      for (int r = 0; r < 8; ++r)
        vreg[nt][r] = vin[(size_t)(b0 + r + 8 * lh) * CAPS_OD + col0 + nt * 16 + li];

    // ---- routing logits: blog += sum_d x_hat * v for this wave's 4 o's ----
    // Tile pair (2t, 2t+1) spans exactly capsule o = 4*wave + t; reduce the
    // 16 lanes of each half with a shfl_xor tree.
    #pragma unroll
    for (int t = 0; t < 4; ++t) {
      #pragma unroll
      for (int r = 0; r < 8; ++r) {
        float p = xh[2 * t][r] * vreg[2 * t][r] + xh[2 * t + 1][r] * vreg[2 * t + 1][r];
        p += __shfl_xor(p, 1, 32);
        p += __shfl_xor(p, 2, 32);
        p += __shfl_xor(p, 4, 32);
        p += __shfl_xor(p, 8, 32);
        if (li == 0) {
          int br = r + 8 * lh;
          int o  = wave * 4 + t;
          size_t gi = ((size_t)(b0 + br) * CAPS_NI + i) * CAPS_NO + o;
          float bn = p + (use_bold ? blog[gi] : 0.0f);
          blog[gi] = bn;                    // in-place safe: 1 writer/elem
          cb[br][o] = bn;
        }
      }
    }
    __syncthreads();

    // softmax over o (32 values) per batch row: 16 rows, 1 thread each
    if (tid < BT) {
      float m = -3.0e38f;
      for (int o = 0; o < CAPS_NO; ++o) m = fmaxf(m, cb[tid][o]);
      float sum = 0.0f;
      for (int o = 0; o < CAPS_NO; ++o) {
        float e = __expf(cb[tid][o] - m);
        cb[tid][o] = e;
        sum += e;
      }
      rinv[tid] = 1.0f / sum;
    }
    __syncthreads();

    // s accumulation: sacc += c[b,o] * x_hat  (c via LDS broadcast)
    #pragma unroll
    for (int nt = 0; nt < 8; ++nt) {
      int o = wave * 4 + (nt >> 1);
      #pragma unroll
      for (int r = 0; r < 8; ++r) {
        int br = r + 8 * lh;
        float c = cb[br][o] * rinv[br];
        sacc[nt][r] += c * xh[nt][r];
      }
    }
  }

  // spill per-chunk partial s (reduced over i later)
  float* sp = s_part + ((size_t)ic * CAPS_B + b0) * CAPS_OD;
  #pragma unroll
  for (int nt = 0; nt < 8; ++nt)
    #pragma unroll
    for (int r = 0; r < 8; ++r)
      sp[(size_t)(r + 8 * lh) * CAPS_OD + col0 + nt * 16 + li] = sacc[nt][r];
}

// s = sum over chunks of s_part; accumulate global squared-norm for squash
__global__ __launch_bounds__(256)
void caps_reduce(const float* __restrict__ s_part, float* __restrict__ s,
                 float* __restrict__ norm_acc)
{
  int idx = blockIdx.x * 256 + threadIdx.x;
  float acc = 0.0f;
  for (int c = 0; c < N_CHUNKS; ++c) acc += s_part[(size_t)c * S_ELEMS + idx];
  s[idx] = acc;

  __shared__ float red[256];
  red[threadIdx.x] = acc * acc;
  __syncthreads();
  for (int off = 128; off > 0; off >>= 1) {
    if (threadIdx.x < off) red[threadIdx.x] += red[threadIdx.x + off];
    __syncthreads();
  }
  if (threadIdx.x == 0) atomicAdd(norm_acc, red[0]);
}

// squash with GLOBAL norm (faithful to reference: tf.norm axis=None)
__global__ __launch_bounds__(256)
void caps_squash(const float* __restrict__ s, const float* __restrict__ norm_acc,
                 float* __restrict__ out)
{
  int idx = blockIdx.x * 256 + threadIdx.x;
  float n2 = *norm_acc;
  float n  = sqrtf(n2);
  float scale = n / (1.0f + n2);
  out[idx] = scale * s[idx];
}

extern "C" void kernel_launch(void* const* d_in, const int* in_sizes, int n_in,
                              void* d_out, int out_size, void* d_ws, size_t ws_size,
                              hipStream_t stream) {
  const float* x = (const float*)d_in[0];   // [128][1152][32]
  const float* W = (const float*)d_in[1];   // [1152][32][32][32]

  // workspace layout (floats): ~57.7 MB total
  float* ws     = (float*)d_ws;
  float* s_part = ws;                                       // 72 * 131072
  float* s      = s_part + (size_t)N_CHUNKS * S_ELEMS;      // 131072
  float* v      = s + S_ELEMS;                              // 131072
  float* blog   = v + S_ELEMS;                              // 128*1152*32
  float* norms  = blog + (size_t)CAPS_B * CAPS_NI * CAPS_NO;// 4

  dim3 pg(N_CHUNKS, NBT);        // 72 x 8 = 576 workgroups, 8 waves each
  dim3 rg(S_ELEMS / 256);        // 512 workgroups

  caps_zero_norms<<<1, 256, 0, stream>>>(norms);

  // iteration 0: c uniform, s0 = (1/32) sum_i x_hat
  caps_pass<<<pg, 256, 0, stream>>>(x, W, v, blog, s_part, 0, 0);
  caps_reduce<<<rg, 256, 0, stream>>>(s_part, s, norms + 0);
  caps_squash<<<rg, 256, 0, stream>>>(s, norms + 0, v);

  // iteration 1: b1 = <x_hat, v0>; c = softmax(b1); s1
  caps_pass<<<pg, 256, 0, stream>>>(x, W, v, blog, s_part, 1, 0);
  caps_reduce<<<rg, 256, 0, stream>>>(s_part, s, norms + 1);
  caps_squash<<<rg, 256, 0, stream>>>(s, norms + 1, v);

  // iteration 2: b2 = b1 + <x_hat, v1>; c = softmax(b2); s2 -> output
  caps_pass<<<pg, 256, 0, stream>>>(x, W, v, blog, s_part, 1, 1);
  caps_reduce<<<rg, 256, 0, stream>>>(s_part, s, norms + 2);
  caps_squash<<<rg, 256, 0, stream>>>(s, norms + 2, (float*)d_out);
}